// GaAN_43568148251378
// MI455X (gfx1250) — compile-verified
//
#include <hip/hip_runtime.h>
#include <math.h>

#define N_NODES 50000
#define N_EDGES 600000
#define HIDDIM  128
#define KD      128          // every GEMM in this net has K = 128
#define KHEADS  8
#define D_A     16
#define D_V     16
#define D_M     64
#define NCLS    40
#define SLOPE   0.1f

#define BS_STRIDE 48         // float2 row stride: 96 banks == 32 mod 64 -> the
                             // two wave halves (K-pair rows k, k+1) hit disjoint
                             // 32-bank sets on ds_load_b64

typedef float v2f __attribute__((ext_vector_type(2)));
typedef float v8f __attribute__((ext_vector_type(8)));

// ---------------------------------------------------------------------------
// f32 WMMA: D(16x16) = A(16x4) * B(4x16) + C   (wave32, CDNA5)
// ---------------------------------------------------------------------------
__device__ __forceinline__ v8f wmma4(v2f a, v2f b, v8f c) {
  return __builtin_amdgcn_wmma_f32_16x16x4_f32(
      /*neg_a=*/false, a, /*neg_b=*/false, b,
      /*c_mod=*/(short)0, c, /*reuse_a=*/false, /*reuse_b=*/false);
}

// C[M,N] = A[M,128] @ B[128,N] (+ Cin) (+ leaky).
// One block = 8 waves = 8 M-tiles sharing one 32-wide N stripe (two 16-col
// tiles per wave -> each A fragment feeds two WMMAs, halving A traffic).
// B staged in LDS as K-pair float2s: element [kp][c] = {B[2kp][c], B[2kp+1][c]}
// so each WMMA B fragment is ONE aligned ds_load_b64 (no repacking movs).
// Out-of-range output columns are clamped on load (lane n's B fragment only
// feeds output column n, which is never stored): hot loop is divergence-free
// and WMMA always runs with EXEC all-1s.
__global__ __launch_bounds__(256) void gemm128_wmma_kernel(
    const float* __restrict__ A, const float* __restrict__ B,
    const float* __restrict__ Cin, float* __restrict__ C,
    int tilesM, int Nn, int stripesN, int leaky)
{
  __shared__ float2 Bs[(KD / 2) * BS_STRIDE];   // 64 x 48 float2 = 24 KB

  int stripe  = blockIdx.x % stripesN;
  int tmBase  = (blockIdx.x / stripesN) << 3;
  int tid     = threadIdx.x;
  int colBase = stripe << 5;

  // ---- cooperative B stage: 64 K-pairs x 32 cols, 8 float2 per thread ----
#pragma unroll
  for (int i = 0; i < 8; ++i) {
    int idx = tid + i * 256;
    int kp  = idx >> 5;                  // K pair 0..63
    int c   = idx & 31;
    int col = colBase + c;
    if (col >= Nn) col = Nn - 1;         // clamp: junk column never stored
    float2 bv;
    bv.x = B[(size_t)(2 * kp)     * Nn + col];
    bv.y = B[(size_t)(2 * kp + 1) * Nn + col];
    Bs[kp * BS_STRIDE + c] = bv;
  }
  __syncthreads();

  int wave = tid >> 5;
  int tm   = tmBase + wave;
  if (tm >= tilesM) return;              // wave-uniform

  int lane = tid & 31;
  int half = lane >> 4;                  // 0: K{0,1}/rows 0-7 ; 1: K{2,3}/rows 8-15
  int l16  = lane & 15;
  int rowA  = (tm << 4) + l16;
  int colB0 = colBase + l16;
  int colB1 = colBase + 16 + l16;
  bool ok0 = (colB0 < Nn), ok1 = (colB1 < Nn);
  int ld0 = ok0 ? colB0 : (Nn - 1);
  int ld1 = ok1 ? colB1 : (Nn - 1);

  v8f acc0 = {}, acc1 = {};
  if (Cin) {
    const float* cb = Cin + (size_t)((tm << 4) + (half << 3)) * Nn;
#pragma unroll
    for (int j = 0; j < 8; ++j) {
      acc0[j] = cb[(size_t)j * Nn + ld0];
      acc1[j] = cb[(size_t)j * Nn + ld1];
    }
  }

  const float2* A2  = (const float2*)(A + (size_t)rowA * KD);
  const float2* bsp = Bs + l16;          // K-pair kp of col l16 at [kp*BS_STRIDE]
#pragma unroll
  for (int kk = 0; kk < KD; kk += 4) {
    float2 av = A2[(kk >> 1) + half];    // A[rowA, kk+2*half .. +1]
    v2f a; a.x = av.x; a.y = av.y;
    int kp = (kk >> 1) + half;           // K-pair index for this half
    float2 b0v = bsp[kp * BS_STRIDE];
    float2 b1v = bsp[kp * BS_STRIDE + 16];
    v2f b0; b0.x = b0v.x; b0.y = b0v.y;
    v2f b1; b1.x = b1v.x; b1.y = b1v.y;
    acc0 = wmma4(a, b0, acc0);
    acc1 = wmma4(a, b1, acc1);
  }

  float* cb = C + (size_t)((tm << 4) + (half << 3)) * Nn;
  if (ok0) {
#pragma unroll
    for (int j = 0; j < 8; ++j) {
      float vv = acc0[j];
      if (leaky) vv = vv > 0.0f ? vv : vv * SLOPE;
      cb[(size_t)j * Nn + colB0] = vv;
    }
  }
  if (ok1) {
#pragma unroll
    for (int j = 0; j < 8; ++j) {
      float vv = acc1[j];
      if (leaky) vv = vv > 0.0f ? vv : vv * SLOPE;
      cb[(size_t)j * Nn + colB1] = vv;
    }
  }
}

// ---------------------------------------------------------------------------
// helpers
// ---------------------------------------------------------------------------
__global__ void fill_kernel(float* __restrict__ p, float v, long long n) {
  long long i = (long long)blockIdx.x * blockDim.x + threadIdx.x;
  if (i < n) p[i] = v;
}

// float atomic-max via signed-max / unsigned-min idiom (native int atomics)
__device__ __forceinline__ void atomicMaxF(float* addr, float val) {
  if (val >= 0.0f) atomicMax((int*)addr, __float_as_int(val));
  else             atomicMin((unsigned int*)addr, __float_as_uint(val));
}

// ---------------------------------------------------------------------------
// edge-phase kernels
// ---------------------------------------------------------------------------
__global__ void edge_score_kernel(const float* __restrict__ q,
                                  const float* __restrict__ kb,
                                  const int* __restrict__ ei,
                                  float* __restrict__ score,
                                  float* __restrict__ smax)
{
  int idx = blockIdx.x * blockDim.x + threadIdx.x;
  if (idx >= N_EDGES * KHEADS) return;
  int e = idx >> 3, kh = idx & 7;
  int s = ei[e], t = ei[N_EDGES + e];
  const float4* qp = (const float4*)(q  + (size_t)t * HIDDIM + kh * D_A);
  const float4* kp = (const float4*)(kb + (size_t)s * HIDDIM + kh * D_A);
  float acc = 0.0f;
#pragma unroll
  for (int i = 0; i < 4; ++i) {
    float4 a = qp[i], b = kp[i];
    acc += a.x * b.x + a.y * b.y + a.z * b.z + a.w * b.w;
  }
  score[idx] = acc;
  atomicMaxF(&smax[(size_t)t * KHEADS + kh], acc);
}

__global__ void edge_exp_kernel(float* __restrict__ score,
                                const float* __restrict__ smax,
                                float* __restrict__ denom,
                                const int* __restrict__ ei)
{
  int idx = blockIdx.x * blockDim.x + threadIdx.x;
  if (idx >= N_EDGES * KHEADS) return;
  int e = idx >> 3, kh = idx & 7;
  int t = ei[N_EDGES + e];
  float ex = expf(score[idx] - smax[(size_t)t * KHEADS + kh]);
  score[idx] = ex;
  atomicAdd(&denom[(size_t)t * KHEADS + kh], ex);
}

__global__ void edge_agg_kernel(const float* __restrict__ score,
                                const float* __restrict__ denom,
                                const float* __restrict__ v,
                                float* __restrict__ agg,
                                const int* __restrict__ ei)
{
  int idx = blockIdx.x * blockDim.x + threadIdx.x;
  if (idx >= N_EDGES * KHEADS) return;
  int e = idx >> 3, kh = idx & 7;
  int s = ei[e], t = ei[N_EDGES + e];
  float al = score[idx] / (denom[(size_t)t * KHEADS + kh] + 1e-16f);
  const float4* vp = (const float4*)(v + (size_t)s * HIDDIM + kh * D_V);
  float* ap = agg + (size_t)t * HIDDIM + kh * D_V;
#pragma unroll
  for (int i = 0; i < 4; ++i) {
    float4 vv = vp[i];
    atomicAdd(&ap[i * 4 + 0], al * vv.x);
    atomicAdd(&ap[i * 4 + 1], al * vv.y);
    atomicAdd(&ap[i * 4 + 2], al * vv.z);
    atomicAdd(&ap[i * 4 + 3], al * vv.w);
  }
}

__global__ void edge_pool_kernel(const float* __restrict__ hc,
                                 const float* __restrict__ m,
                                 float* __restrict__ msum,
                                 float* __restrict__ mpool,
                                 float* __restrict__ deg,
                                 const int* __restrict__ ei)
{
  int idx = blockIdx.x * blockDim.x + threadIdx.x;
  if (idx >= N_EDGES * HIDDIM) return;
  int e = idx >> 7, d = idx & 127;
  int s = ei[e], t = ei[N_EDGES + e];
  atomicAdd(&msum[(size_t)t * HIDDIM + d], hc[(size_t)s * HIDDIM + d]);
  if (d < D_M) atomicMaxF(&mpool[(size_t)t * D_M + d], m[(size_t)s * D_M + d]);
  if (d == 0)  atomicAdd(&deg[t], 1.0f);
}

__global__ void gate_kernel(const float* __restrict__ hc,
                            const float* __restrict__ mpool,
                            const float* __restrict__ msum,
                            const float* __restrict__ deg,
                            const float* __restrict__ agg,
                            const float* __restrict__ Wg,
                            float* __restrict__ gated)
{
  int idx = blockIdx.x * blockDim.x + threadIdx.x;
  if (idx >= N_NODES * KHEADS) return;
  int n = idx >> 3, kh = idx & 7;
  const float* hr = hc + (size_t)n * HIDDIM;
  float dg = fmaxf(deg[n], 1.0f);
  float acc = 0.0f;
#pragma unroll 4
  for (int j = 0; j < HIDDIM; ++j) acc += hr[j] * Wg[j * KHEADS + kh];
#pragma unroll 4
  for (int j = 0; j < D_M; ++j) {
    float mv = mpool[(size_t)n * D_M + j];
    mv = (mv < -3.0e38f) ? 0.0f : mv;            // empty segment -> 0
    acc += mv * Wg[(HIDDIM + j) * KHEADS + kh];
  }
#pragma unroll 4
  for (int j = 0; j < HIDDIM; ++j)
    acc += (msum[(size_t)n * HIDDIM + j] / dg) * Wg[(HIDDIM + D_M + j) * KHEADS + kh];
  float g = 1.0f / (1.0f + expf(-acc));
  float* gp = gated + (size_t)n * HIDDIM + kh * D_V;
  const float* ap = agg + (size_t)n * HIDDIM + kh * D_V;
#pragma unroll
  for (int d = 0; d < D_V; ++d) gp[d] = g * ap[d];
}

__global__ void log_softmax_kernel(const float* __restrict__ logits,
                                   float* __restrict__ out)
{
  int n = blockIdx.x * blockDim.x + threadIdx.x;
  if (n >= N_NODES) return;
  const float* r = logits + (size_t)n * NCLS;
  float mx = -3.4e38f;
  for (int j = 0; j < NCLS; ++j) mx = fmaxf(mx, r[j]);
  float s = 0.0f;
  for (int j = 0; j < NCLS; ++j) s += expf(r[j] - mx);
  float ls = logf(s);
  for (int j = 0; j < NCLS; ++j) out[(size_t)n * NCLS + j] = r[j] - mx - ls;
}

// ---------------------------------------------------------------------------
// driver
// ---------------------------------------------------------------------------
extern "C" void kernel_launch(void* const* d_in, const int* in_sizes, int n_in,
                              void* d_out, int out_size, void* d_ws, size_t ws_size,
                              hipStream_t stream)
{
  (void)in_sizes; (void)n_in; (void)out_size; (void)ws_size;
  const float* x    = (const float*)d_in[0];
  const int*   ei   = (const int*)d_in[1];   // [2,E]
  const float* Win  = (const float*)d_in[2];
  const float* Wout = (const float*)d_in[3];
  const float* Wq   = (const float*)d_in[4];
  const float* Wk   = (const float*)d_in[5];
  const float* Wv   = (const float*)d_in[6];
  const float* Wm   = (const float*)d_in[7];
  const float* Wg   = (const float*)d_in[8];
  const float* Wo   = (const float*)d_in[9];
  float* out = (float*)d_out;

  const size_t NH = (size_t)N_NODES * HIDDIM;
  float* p = (float*)d_ws;
  float* h0    = p; p += NH;
  float* h1    = p; p += NH;
  float* q     = p; p += NH;                 // reused as `gated` after attention
  float* kb    = p; p += NH;                 // reused as `tmp` for Wo partial
  float* v     = p; p += NH;
  float* agg   = p; p += NH;
  float* msum  = p; p += NH;
  float* m     = p; p += (size_t)N_NODES * D_M;
  float* mpool = p; p += (size_t)N_NODES * D_M;
  float* smax  = p; p += (size_t)N_NODES * KHEADS;
  float* denom = p; p += (size_t)N_NODES * KHEADS;
  float* deg   = p; p += N_NODES;
  float* score = p; p += (size_t)N_EDGES * KHEADS;
  float* logits= p; p += (size_t)N_NODES * NCLS;

  auto gemm = [&](const float* A, const float* B, const float* Cin, float* C,
                  int M, int Nn, int leaky) {
    int tilesM   = M / 16;
    int stripesN = (Nn + 31) / 32;
    int blocks   = ((tilesM + 7) / 8) * stripesN;
    gemm128_wmma_kernel<<<blocks, 256, 0, stream>>>(A, B, Cin, C,
                                                    tilesM, Nn, stripesN, leaky);
  };
  auto fill = [&](float* ptr, float val, long long n) {
    fill_kernel<<<(unsigned)((n + 255) / 256), 256, 0, stream>>>(ptr, val, n);
  };

  // h = x @ W_in
  gemm(x, Win, nullptr, h0, N_NODES, HIDDIM, 0);

  const int EK = N_EDGES * KHEADS;
  const int EH = N_EDGES * HIDDIM;

  for (int L = 0; L < 2; ++L) {
    const float* hc = (L == 0) ? h0 : h1;
    float*       hn = (L == 0) ? h1 : h0;
    const float* wq = Wq + (size_t)L * HIDDIM * (KHEADS * D_A);
    const float* wk = Wk + (size_t)L * HIDDIM * (KHEADS * D_A);
    const float* wv = Wv + (size_t)L * HIDDIM * (KHEADS * D_V);
    const float* wm = Wm + (size_t)L * HIDDIM * D_M;
    const float* wg = Wg + (size_t)L * (HIDDIM + D_M + HIDDIM) * KHEADS;
    const float* wo = Wo + (size_t)L * (HIDDIM + KHEADS * D_V) * HIDDIM;

    fill(smax, -INFINITY, (long long)N_NODES * KHEADS);
    fill(denom, 0.0f,     (long long)N_NODES * KHEADS);
    fill(agg,   0.0f,     (long long)NH);
    fill(mpool, -INFINITY,(long long)N_NODES * D_M);
    fill(msum,  0.0f,     (long long)NH);
    fill(deg,   0.0f,     (long long)N_NODES);

    gemm(hc, wq, nullptr, q,  N_NODES, HIDDIM, 0);
    gemm(hc, wk, nullptr, kb, N_NODES, HIDDIM, 0);
    gemm(hc, wv, nullptr, v,  N_NODES, HIDDIM, 0);
    gemm(hc, wm, nullptr, m,  N_NODES, D_M,    0);

    edge_score_kernel<<<(EK + 255) / 256, 256, 0, stream>>>(q, kb, ei, score, smax);
    edge_exp_kernel  <<<(EK + 255) / 256, 256, 0, stream>>>(score, smax, denom, ei);
    edge_agg_kernel  <<<(EK + 255) / 256, 256, 0, stream>>>(score, denom, v, agg, ei);
    edge_pool_kernel <<<(EH + 255) / 256, 256, 0, stream>>>(hc, m, msum, mpool, deg, ei);
    gate_kernel<<<((N_NODES * KHEADS) + 255) / 256, 256, 0, stream>>>(
        hc, mpool, msum, deg, agg, wg, /*gated=*/q);

    // concat([h, gated]) @ Wo  ->  two K=128 GEMMs, leaky fused into the second
    gemm(hc, wo,                   nullptr, kb, N_NODES, HIDDIM, 0);
    gemm(q,  wo + HIDDIM * HIDDIM, kb,      hn, N_NODES, HIDDIM, 1);
  }

  gemm(h0, Wout, nullptr, logits, N_NODES, NCLS, 0);   // h0 = output of layer 2
  log_softmax_kernel<<<(N_NODES + 255) / 256, 256, 0, stream>>>(logits, out);
}